// InteractionBlock_5016521802056
// MI455X (gfx1250) — compile-verified
//
#include <hip/hip_runtime.h>

typedef __attribute__((ext_vector_type(2))) float v2f;
typedef __attribute__((ext_vector_type(8))) float v8f;

#define B_ 2
#define G_ 50000
#define K_ 32
#define A_ 64

#define WAVES_PER_BLOCK 8
#define TILE_M 16
#define MSG_STRIDE 68    // A-frag ds_load_b64 conflict-free
#define WPAIR_STRIDE 160 // k-pair row stride (words); +160 mod 64 = +32 -> halves hit disjoint banks

__global__ __launch_bounds__(256)
void interaction_block_kernel(const float* __restrict__ out_in,
                              const int*   __restrict__ nbr_idx,
                              const float* __restrict__ W2g,
                              const float* __restrict__ b2g,
                              const float* __restrict__ W3g,
                              const float* __restrict__ b3g,
                              float*       __restrict__ y_out)
{
    // W stored k-pair interleaved: sW[(k>>1)*WPAIR_STRIDE + 2n + (k&1)]
    // -> each B-fragment (W[k0][n], W[k0+1][n]) is one aligned ds_load_b64.
    __shared__ float sW2[(A_ / 2) * WPAIR_STRIDE];
    __shared__ float sW3[(A_ / 2) * WPAIR_STRIDE];
    __shared__ float sb2[A_];
    __shared__ float sb3[A_];
    __shared__ float sMsg[WAVES_PER_BLOCK][TILE_M * MSG_STRIDE];

    const int tid = threadIdx.x;

    // Stage weights/biases once per block (interleave k-pairs on the fly).
    for (int i = tid; i < A_ * A_; i += 256) {
        const int k = i >> 6;         // row (k-dim)
        const int n = i & 63;         // col (n-dim)
        const int s = (k >> 1) * WPAIR_STRIDE + 2 * n + (k & 1);
        sW2[s] = W2g[i];
        sW3[s] = W3g[i];
    }
    if (tid < A_) { sb2[tid] = b2g[tid]; sb3[tid] = b3g[tid]; }
    __syncthreads();

    const int wave = tid >> 5;
    const int lane = tid & 31;
    const int tile = blockIdx.x * WAVES_PER_BLOCK + wave;
    const int n_tiles = (B_ * G_) / TILE_M;   // 6250
    if (tile >= n_tiles) return;

    const int row0 = tile * TILE_M;           // flat row in [0, B*G)
    const int b    = row0 / G_;               // tile never crosses batch edge
    const int g0   = row0 % G_;
    float* msg = sMsg[wave];

    // ---------------- Phase 1: neighbor gather-sum into LDS tile ----------
    // Lane l owns columns {2l, 2l+1}. Indices broadcast via readlane so the
    // padding test is a scalar branch and loads use SGPR base + lane offset.
    const float* outB   = out_in + (size_t)b * G_ * A_;
    const float* lanePt = outB + 2 * lane;
    for (int m = 0; m < TILE_M; ++m) {
        const int g = g0 + m;
        const int myidx = nbr_idx[g * K_ + lane];   // coalesced 128B
        float ax = 0.f, ay = 0.f;
        #pragma unroll
        for (int k = 0; k < K_; ++k) {
            const int idx = __builtin_amdgcn_readlane(myidx, k); // wave-uniform SGPR
            if (idx < G_) {                          // idx==G_ -> zero pad row
                const v2f val = *(const v2f*)(lanePt + (size_t)idx * A_);
                ax += val.x;
                ay += val.y;
            }
        }
        msg[m * MSG_STRIDE + 2 * lane]     = ax;
        msg[m * MSG_STRIDE + 2 * lane + 1] = ay;
    }

    // Fragment coordinate helpers (wave32 WMMA layouts, ISA 7.12.2):
    //  A 16x4 f32 : lane m (0-15) -> (m, k0..k0+1); lanes 16-31 -> k0+2..k0+3
    //  B 4x16     : lane l -> rows {k0+2*(l>>4), +1}, col = l&15  (pair load)
    //  C/D 16x16  : vgpr v -> row v + 8*(lane>=16), col = lane&15
    const int am    = lane & 15;
    const int khalf = lane >> 4;          // selects k-pair within the 4-wide block
    const int ncol  = lane & 15;
    const int rsel  = (lane >> 4) * 8;

    const float* aBase = msg + am * MSG_STRIDE + 2 * khalf;     // + kb*4
    // ---------------- Phase 2: T = msg @ W2 + b2 + out --------------------
    v8f acc[4];
    #pragma unroll
    for (int nb = 0; nb < 4; ++nb) {
        v8f c = {};
        const int n = nb * 16 + ncol;
        const float* bBase = sW2 + khalf * WPAIR_STRIDE + 2 * n; // + kb*2*WPAIR_STRIDE
        #pragma unroll
        for (int kb = 0; kb < 16; ++kb) {
            const v2f a = *(const v2f*)(aBase + kb * 4);
            const v2f w = *(const v2f*)(bBase + kb * 2 * WPAIR_STRIDE);
            c = __builtin_amdgcn_wmma_f32_16x16x4_f32(
                    false, a, false, w, (short)0, c, false, false);
        }
        acc[nb] = c;
    }

    // bias + residual, using C/D lane mapping
    #pragma unroll
    for (int nb = 0; nb < 4; ++nb) {
        const int n = nb * 16 + ncol;
        #pragma unroll
        for (int v = 0; v < 8; ++v) {
            const int row = v + rsel;
            acc[nb][v] += sb2[n] + out_in[((size_t)row0 + row) * A_ + n];
        }
    }

    // write T back over the wave-private LDS tile (DS ops in-order per wave)
    #pragma unroll
    for (int nb = 0; nb < 4; ++nb) {
        const int n = nb * 16 + ncol;
        #pragma unroll
        for (int v = 0; v < 8; ++v) {
            msg[(v + rsel) * MSG_STRIDE + n] = acc[nb][v];
        }
    }

    // ---------------- Phase 3: Y = T @ W3 + b3, store ---------------------
    #pragma unroll
    for (int nb = 0; nb < 4; ++nb) {
        v8f c = {};
        const int n = nb * 16 + ncol;
        const float* bBase = sW3 + khalf * WPAIR_STRIDE + 2 * n;
        #pragma unroll
        for (int kb = 0; kb < 16; ++kb) {
            const v2f a = *(const v2f*)(aBase + kb * 4);
            const v2f w = *(const v2f*)(bBase + kb * 2 * WPAIR_STRIDE);
            c = __builtin_amdgcn_wmma_f32_16x16x4_f32(
                    false, a, false, w, (short)0, c, false, false);
        }
        #pragma unroll
        for (int v = 0; v < 8; ++v) {
            const int row = v + rsel;
            y_out[((size_t)row0 + row) * A_ + n] = c[v] + sb3[n];
        }
    }
}

extern "C" void kernel_launch(void* const* d_in, const int* in_sizes, int n_in,
                              void* d_out, int out_size, void* d_ws, size_t ws_size,
                              hipStream_t stream) {
    const float* outp = (const float*)d_in[0];
    const int*   idx  = (const int*)d_in[1];
    const float* W2   = (const float*)d_in[2];
    const float* b2   = (const float*)d_in[3];
    const float* W3   = (const float*)d_in[4];
    const float* b3   = (const float*)d_in[5];
    float* y = (float*)d_out;

    const int n_tiles = (B_ * G_) / TILE_M;                       // 6250
    const int blocks  = (n_tiles + WAVES_PER_BLOCK - 1) / WAVES_PER_BLOCK;

    hipLaunchKernelGGL(interaction_block_kernel, dim3(blocks), dim3(256), 0, stream,
                       outp, idx, W2, b2, W3, b3, y);
}